// Enhancement_77309412162
// MI455X (gfx1250) — compile-verified
//
#include <hip/hip_runtime.h>
#include <hip/hip_bf16.h>

typedef __attribute__((ext_vector_type(2))) float v2f;
typedef __attribute__((ext_vector_type(8))) float v8f;

#define CC 512
#define NN 2304            // 48*48
#define BB 8
#define TILES_PER_B (NN / 16)   // 144

// LDS weight staging: [g_w(512) | theta_w(512) | phi_w(512) | zeros(512)]
#define WOFF_G    0
#define WOFF_TH   512
#define WOFF_PH   1024
#define WOFF_ZERO 1536
#define WLDS_SZ   2048

// ---------------------------------------------------------------------------
// Kernel 1: rank-1 projections via V_WMMA_F32_16X16X4_F32.
// A (16x4, MxK) = weight rows (row0=g_w,row1=theta_w | row0=phi_w), rest zero.
// B (4x16, KxN) = 4 channels x 16 contiguous spatial positions (coalesced).
// Weights pre-packed in LDS so the K-loop has NO divergent control flow:
// each lane reads its A-fragment with one unconditional ds_load_b64 per chain
// (rows >= 2 read from a zero region).
// ---------------------------------------------------------------------------
__global__ __launch_bounds__(128) void proj_kernel(
    const float* __restrict__ x, const float* __restrict__ x1,
    const float* __restrict__ x2,
    const float* __restrict__ g_w, const float* __restrict__ th_w,
    const float* __restrict__ ph_w,
    const float* __restrict__ g_b, const float* __restrict__ th_b,
    const float* __restrict__ ph_b,
    float* __restrict__ g_o, float* __restrict__ th_o,
    float* __restrict__ p1_o, float* __restrict__ p2_o)
{
    __shared__ float wlds[WLDS_SZ];

    // Stage weights + zero pad (128 threads, 4 rounds each region)
    for (int i = threadIdx.x; i < CC; i += 128) {
        wlds[WOFF_G    + i] = g_w[i];
        wlds[WOFF_TH   + i] = th_w[i];
        wlds[WOFF_PH   + i] = ph_w[i];
        wlds[WOFF_ZERO + i] = 0.f;
    }
    __syncthreads();

    const int wave = (blockIdx.x * blockDim.x + threadIdx.x) >> 5;
    const int lane = threadIdx.x & 31;
    const int b    = wave / TILES_PER_B;
    const int tp   = (wave % TILES_PER_B) * 16;
    const int m    = lane & 15;          // M (weight row) / N (position)
    const int ko   = (lane >> 4) << 1;   // K offset: 0 (lanes 0-15), 2 (16-31)
    const int pos  = tp + m;

    // Loop-invariant per-lane weight bases (rows >= 2 -> zero region)
    const int baseA_x = (m == 0) ? WOFF_G  : (m == 1) ? WOFF_TH : WOFF_ZERO;
    const int baseA_p = (m == 0) ? WOFF_PH : WOFF_ZERO;

    const float* xb  = x  + (size_t)b * CC * NN + pos;
    const float* x1b = x1 + (size_t)b * CC * NN + pos;
    const float* x2b = x2 + (size_t)b * CC * NN + pos;

    v8f accx = {};  // row0 -> g proj, row1 -> theta proj
    v8f acc1 = {};  // row0 -> phi(x1)
    v8f acc2 = {};  // row0 -> phi(x2)

#pragma unroll 2
    for (int c = 0; c < CC; c += 4) {
        // A operands from LDS: unconditional 8-byte reads
        const v2f aw = *(const __shared__ v2f*)&wlds[baseA_x + c + ko];
        const v2f ap = *(const __shared__ v2f*)&wlds[baseA_p + c + ko];

        // B operands (data): lane -> position, half-wave -> K pair
        const size_t base = (size_t)(c + ko) * NN;
        v2f bx; bx.x = xb [base]; bx.y = xb [base + NN];
        v2f b1; b1.x = x1b[base]; b1.y = x1b[base + NN];
        v2f b2; b2.x = x2b[base]; b2.y = x2b[base + NN];

        accx = __builtin_amdgcn_wmma_f32_16x16x4_f32(false, aw, false, bx,
                                                     (short)0, accx, false, false);
        acc1 = __builtin_amdgcn_wmma_f32_16x16x4_f32(false, ap, false, b1,
                                                     (short)0, acc1, false, false);
        acc2 = __builtin_amdgcn_wmma_f32_16x16x4_f32(false, ap, false, b2,
                                                     (short)0, acc2, false, false);
    }

    // D layout: VGPR v, lanes 0-15 hold row M=v. Rows 0/1 live in lanes 0-15.
    if (lane < 16) {
        const int o = b * NN + pos;
        g_o [o] = accx[0] + g_b[0];
        th_o[o] = accx[1] + th_b[0];
        p1_o[o] = acc1[0] + ph_b[0];
        p2_o[o] = acc2[0] + ph_b[0];
    }
}

// ---------------------------------------------------------------------------
// Kernel 2: per-batch reductions: s1=<phi1,g>, s2=<phi2,g>, ts=sum(theta),
// tq=sum(theta^2). One block per batch.
// ---------------------------------------------------------------------------
__global__ __launch_bounds__(256) void stats_kernel(
    const float* __restrict__ g, const float* __restrict__ th,
    const float* __restrict__ p1, const float* __restrict__ p2,
    float* __restrict__ stats)
{
    __shared__ float sm[4][256];
    const int b = blockIdx.x, tid = threadIdx.x;
    float s1 = 0.f, s2 = 0.f, ts = 0.f, tq = 0.f;
    for (int i = tid; i < NN; i += 256) {
        const int o = b * NN + i;
        const float gv = g[o], tv = th[o];
        s1 += p1[o] * gv;
        s2 += p2[o] * gv;
        ts += tv;
        tq += tv * tv;
    }
    sm[0][tid] = s1; sm[1][tid] = s2; sm[2][tid] = ts; sm[3][tid] = tq;
    __syncthreads();
    for (int s = 128; s > 0; s >>= 1) {
        if (tid < s)
            for (int k = 0; k < 4; ++k) sm[k][tid] += sm[k][tid + s];
        __syncthreads();
    }
    if (tid < 4) stats[b * 4 + tid] = sm[tid][0];
}

// ---------------------------------------------------------------------------
// Kernel 3: tiny -- coef[b], ybar, Var[y], per-channel alpha. Single block.
// ---------------------------------------------------------------------------
__global__ __launch_bounds__(512) void coef_kernel(
    const float* __restrict__ stats, const float* __restrict__ a,
    const float* __restrict__ bsc, const float* __restrict__ Ww,
    const float* __restrict__ bnw,
    float* __restrict__ coef_o, float* __restrict__ ybar_o,
    float* __restrict__ alpha_o)
{
    float ac = a[0];
    ac = fminf(fmaxf(ac, 0.f), 1.f);
    const float bb = bsc[0];
    const float invN = 1.f / (float)NN;

    float cf[BB];
    float ysum = 0.f, y2 = 0.f;
    for (int i = 0; i < BB; ++i) {
        const float s1 = stats[i * 4 + 0];
        const float s2 = stats[i * 4 + 1];
        const float ts = stats[i * 4 + 2];
        const float tq = stats[i * 4 + 3];
        const float c = bb * invN * (ac * s1 + (1.f - ac) * s2);
        cf[i] = c;
        ysum += c * ts;       // sum of y over batch b
        y2   += c * c * tq;   // sum of y^2 over batch b
    }
    const float invBN = 1.f / (float)(BB * NN);
    const float ybar = ysum * invBN;
    const float vy   = y2 * invBN - ybar * ybar;

    const int tid = threadIdx.x;
    if (tid < BB) coef_o[tid] = cf[tid];
    if (tid == 0) ybar_o[0] = ybar;
    if (tid < CC) {
        const float w = Ww[tid];
        alpha_o[tid] = w * rsqrtf(w * w * vy + 1e-5f) * bnw[tid];
    }
}

// ---------------------------------------------------------------------------
// Kernel 4: bandwidth-dominant elementwise pass.
// out[b,c,hw] = alpha[c]*(theta[b,hw]*coef[b] - ybar) + bn_b[c] + x[b,c,hw]
// One (b,c) row (2304 floats) per block; float4 vectorized.
// ---------------------------------------------------------------------------
__global__ __launch_bounds__(288) void final_kernel(
    const float* __restrict__ x, const float* __restrict__ th,
    const float* __restrict__ coef, const float* __restrict__ ybar_p,
    const float* __restrict__ alpha, const float* __restrict__ bnb,
    float* __restrict__ out)
{
    const int bc = blockIdx.x;
    const int b  = bc >> 9;     // / 512
    const int c  = bc & 511;
    const float al = alpha[c];
    const float be = bnb[c];
    const float cf = coef[b];
    const float yb = ybar_p[0];

    const float4* xr  = (const float4*)(x  + (size_t)bc * NN);
    const float4* tr  = (const float4*)(th + (size_t)b  * NN);
    float4*       orw = (float4*)(out + (size_t)bc * NN);

    const int t = threadIdx.x;
#pragma unroll
    for (int k = 0; k < 2; ++k) {
        const int i = t + k * 288;   // 576 float4 per row
        const float4 vx = xr[i];
        const float4 vt = tr[i];
        float4 r;
        r.x = al * (vt.x * cf - yb) + be + vx.x;
        r.y = al * (vt.y * cf - yb) + be + vx.y;
        r.z = al * (vt.z * cf - yb) + be + vx.z;
        r.w = al * (vt.w * cf - yb) + be + vx.w;
        orw[i] = r;
    }
}

// ---------------------------------------------------------------------------
extern "C" void kernel_launch(void* const* d_in, const int* in_sizes, int n_in,
                              void* d_out, int out_size, void* d_ws, size_t ws_size,
                              hipStream_t stream)
{
    (void)in_sizes; (void)n_in; (void)out_size; (void)ws_size;

    const float* x    = (const float*)d_in[0];
    const float* x1   = (const float*)d_in[1];
    const float* x2   = (const float*)d_in[2];
    const float* g_w  = (const float*)d_in[3];
    const float* g_b  = (const float*)d_in[4];
    const float* th_w = (const float*)d_in[5];
    const float* th_b = (const float*)d_in[6];
    const float* ph_w = (const float*)d_in[7];
    const float* ph_b = (const float*)d_in[8];
    const float* W_w  = (const float*)d_in[9];
    // d_in[10] = W_b: cancels exactly in BatchNorm, unused.
    const float* bn_w = (const float*)d_in[11];
    const float* bn_b = (const float*)d_in[12];
    const float* a    = (const float*)d_in[13];
    const float* bsc  = (const float*)d_in[14];
    float* out = (float*)d_out;

    // Workspace layout (floats)
    float* ws    = (float*)d_ws;
    float* g_o   = ws;                       // 18432
    float* th_o  = g_o  + BB * NN;           // 18432
    float* p1_o  = th_o + BB * NN;           // 18432
    float* p2_o  = p1_o + BB * NN;           // 18432
    float* stats = p2_o + BB * NN;           // 32
    float* coef  = stats + BB * 4;           // 8
    float* ybar  = coef + BB;                // 2
    float* alpha = ybar + 2;                 // 512

    // 1) projections: 8*144 = 1152 waves, 4 waves (128 thr) per block
    proj_kernel<<<(BB * TILES_PER_B) / 4, 128, 0, stream>>>(
        x, x1, x2, g_w, th_w, ph_w, g_b, th_b, ph_b, g_o, th_o, p1_o, p2_o);

    // 2) per-batch reductions
    stats_kernel<<<BB, 256, 0, stream>>>(g_o, th_o, p1_o, p2_o, stats);

    // 3) scalars + per-channel alpha
    coef_kernel<<<1, 512, 0, stream>>>(stats, a, bsc, W_w, bn_w, coef, ybar, alpha);

    // 4) fused elementwise output
    final_kernel<<<BB * CC, 288, 0, stream>>>(x, th_o, coef, ybar, alpha, bn_b, out);
}